// stl_decomp_62483184222298
// MI455X (gfx1250) — compile-verified
//
#include <hip/hip_runtime.h>
#include <hip/hip_bf16.h>

// ---------------------------------------------------------------------------
// STL(15) on [128, 480, 256] f32. Non-robust STL is linear in x:
//   out0 = x - trend, out1 = trend, trend(b,:,c) = R @ x(b,:,c)
// where R (480x480) = M_trend (G + S1 M_trend G), G = I - S1,
//   S1 = (P - M_lp*Conv) * E   (E = per-phase M_seas operator).
// R is built on-device each call (deterministic), then one WMMA f32 GEMM
// R(480x480) x X_b(480x256) per batch writes both outputs in [B,T,C] order.
//
// Main kernel: 32(t) x 256(c) per block, 8 waves, each wave = 2 t-tiles x
// 2 c-tiles of 16x16 accumulated with V_WMMA_F32_16X16X4_F32. B registers are
// shared across the two t-tiles -> 4 wmma : 4 vmem : 2 ds per K=4 step.
// ---------------------------------------------------------------------------

typedef __attribute__((ext_vector_type(2))) float v2f;
typedef __attribute__((ext_vector_type(8))) float v8f;

#define T_LEN   480
#define C_LEN   256
#define B_LEN   128
#define PERIODN 15
#define NSUB    32
#define EXTROWS 34   /* NSUB + 2 */
#define EXTLEN  510  /* EXTROWS * PERIODN */

// ---- workspace layout (in floats) ----
static const size_t OFF_MLP   = 0;
static const size_t OFF_MTR   = OFF_MLP   + (size_t)T_LEN * T_LEN;
static const size_t OFF_MSEAS = OFF_MTR   + (size_t)T_LEN * T_LEN;   // 34*32
static const size_t OFF_WCONV = OFF_MSEAS + (size_t)EXTROWS * NSUB;  // 31
static const size_t OFF_E     = OFF_WCONV + 32;
static const size_t OFF_F     = OFF_E     + (size_t)EXTLEN * T_LEN;
static const size_t OFF_LPF   = OFF_F     + (size_t)T_LEN * T_LEN;
static const size_t OFF_S1    = OFF_LPF   + (size_t)T_LEN * T_LEN;
static const size_t OFF_G     = OFF_S1    + (size_t)T_LEN * T_LEN;
static const size_t OFF_H     = OFF_G     + (size_t)T_LEN * T_LEN;
static const size_t OFF_K     = OFF_H     + (size_t)T_LEN * T_LEN;
static const size_t OFF_R     = OFF_K     + (size_t)T_LEN * T_LEN;
// total ~2.32M floats ~= 9.3 MB of d_ws

// ---------------------------------------------------------------------------
// LOESS smoothing matrix rows (degree-1, tricube), matching _loess_matrix.
// ---------------------------------------------------------------------------
__global__ void stl_build_loess(float* __restrict__ M, int n, int q, int t0, int rows) {
  int r = blockIdx.x * blockDim.x + threadIdx.x;
  if (r >= rows) return;
  int t = t0 + r;
  float* row = M + (size_t)r * n;
  for (int i = 0; i < n; ++i) row[i] = 0.0f;

  int qq = (q < n) ? q : n;
  int left = t - (q - 1) / 2;
  if (left < 0) left = 0;
  if (left > n - qq) left = n - qq;
  int right = left + qq - 1;
  int hi = (t - left) > (right - t) ? (t - left) : (right - t);
  double h = (double)hi;
  if (q > n) h += (double)(q - n) * 0.5;

  double w[32];
  double wsum = 0.0;
  for (int i = 0; i < qq; ++i) {
    double dd = fabs((double)(left + i - t)) / h;
    double u = 1.0 - dd * dd * dd;
    if (u < 0.0) u = 0.0;
    double ww = u * u * u;
    w[i] = ww;
    wsum += ww;
  }
  double xbar = 0.0;
  for (int i = 0; i < qq; ++i) { w[i] /= wsum; xbar += w[i] * (double)(left + i); }
  double var = 0.0;
  for (int i = 0; i < qq; ++i) {
    double dx = (double)(left + i) - xbar;
    var += w[i] * dx * dx;
  }
  for (int i = 0; i < qq; ++i) {
    double a = w[i];
    if (var > 1e-12) {
      double dx = (double)(left + i) - xbar;
      a = w[i] * (1.0 + ((double)t - xbar) * dx / var);
    }
    row[left + i] = (float)a;
  }
}

// E (510x480): E[s*15+j, n*15+j] = M_seas[s, n]
__global__ void stl_build_E(const float* __restrict__ Mseas, float* __restrict__ E) {
  int idx = blockIdx.x * blockDim.x + threadIdx.x;
  if (idx >= EXTLEN * T_LEN) return;
  int m = idx / T_LEN;
  int d = idx - m * T_LEN;
  int j = m % PERIODN;
  int s = m / PERIODN;
  int jp = d % PERIODN;
  int nn = d / PERIODN;
  E[idx] = (j == jp) ? Mseas[s * NSUB + nn] : 0.0f;
}

// combined MA(15)∘MA(15)∘MA(3) forward-window conv weights, length 31
__global__ void stl_build_convw(float* __restrict__ wout) {
  if (threadIdx.x != 0 || blockIdx.x != 0) return;
  float u[29];
  for (int i = 0; i < 29; ++i) {
    int v = i + 1;
    if (29 - i < v) v = 29 - i;
    if (v > 15) v = 15;
    u[i] = (float)v;                 // box15 (x) box15
  }
  for (int k = 0; k < 31; ++k) {
    float s = 0.0f;
    for (int j = 0; j < 3; ++j) {
      int ii = k - j;
      if (ii >= 0 && ii < 29) s += u[ii];
    }
    wout[k] = s / 675.0f;            // / (15*15*3)
  }
}

// F[t,d] = sum_k w[k] * E[t+k, d]    (F = Λ·E, 480x480)
__global__ void stl_conv_F(const float* __restrict__ E, const float* __restrict__ w,
                           float* __restrict__ F) {
  int idx = blockIdx.x * blockDim.x + threadIdx.x;
  if (idx >= T_LEN * T_LEN) return;
  int t = idx / T_LEN;
  int d = idx - t * T_LEN;
  float s = 0.0f;
  #pragma unroll
  for (int k = 0; k < 31; ++k) s += w[k] * E[(size_t)(t + k) * T_LEN + d];
  F[idx] = s;
}

// S1[t,d] = E[15+t, d] - LpF[t,d];  G = I - S1
__global__ void stl_S1G(const float* __restrict__ E, const float* __restrict__ LpF,
                        float* __restrict__ S1, float* __restrict__ G) {
  int idx = blockIdx.x * blockDim.x + threadIdx.x;
  if (idx >= T_LEN * T_LEN) return;
  int t = idx / T_LEN;
  int d = idx - t * T_LEN;
  float s1 = E[(size_t)(PERIODN + t) * T_LEN + d] - LpF[idx];
  S1[idx] = s1;
  G[idx] = ((t == d) ? 1.0f : 0.0f) - s1;
}

// simple tiled f32 matmul, 480x480: C = A*B (+ Cin)   -- setup-only (~0.9 GFLOP)
__global__ void stl_mm480(const float* __restrict__ A, const float* __restrict__ B,
                          const float* __restrict__ Cin, float* __restrict__ C) {
  __shared__ float As[16][17];
  __shared__ float Bs[16][17];
  int tx = threadIdx.x, ty = threadIdx.y;
  int row = blockIdx.y * 16 + ty;
  int col = blockIdx.x * 16 + tx;
  float acc = 0.0f;
  for (int k0 = 0; k0 < T_LEN; k0 += 16) {
    As[ty][tx] = A[(size_t)row * T_LEN + k0 + tx];
    Bs[ty][tx] = B[(size_t)(k0 + ty) * T_LEN + col];
    __syncthreads();
    #pragma unroll
    for (int kk = 0; kk < 16; ++kk) acc += As[ty][kk] * Bs[kk][tx];
    __syncthreads();
  }
  size_t o = (size_t)row * T_LEN + col;
  C[o] = acc + (Cin ? Cin[o] : 0.0f);
}

// ---------------------------------------------------------------------------
// Main GEMM. Block = (one batch b, 32-row t-strip). 8 waves x (2 t-tiles x
// 2 c-tiles). R strip transposed in LDS as two [k][i] strips, stride 16:
// WMMA A-reads are conflict-free (lanes 0-15 banks k*16+0..15, lanes 16-31 at
// +32 since K-offset 2 * stride 16 = 32 words).
// ---------------------------------------------------------------------------
__global__ __launch_bounds__(256) void stl_trend_wmma(const float* __restrict__ x,
                                                      const float* __restrict__ R,
                                                      float* __restrict__ out_sr,
                                                      float* __restrict__ out_tr) {
  __shared__ float A0[T_LEN * 16];   // rows t0   .. t0+15, [k][i]
  __shared__ float A1[T_LEN * 16];   // rows t0+16.. t0+31, [k][i]
  const int t0 = blockIdx.x * 32;
  const int b  = blockIdx.y;
  const int tid = threadIdx.x;

  for (int idx = tid; idx < 32 * T_LEN; idx += 256) {
    int i = idx / T_LEN;           // strip row (0..31)
    int k = idx - i * T_LEN;       // column of R (coalesced over k)
    float v = R[(size_t)(t0 + i) * T_LEN + k];
    if (i < 16) A0[k * 16 + i] = v;
    else        A1[k * 16 + (i - 16)] = v;
  }
  __syncthreads();

  const int wave = tid >> 5;
  const int lane = tid & 31;
  const int lo   = lane & 15;
  const int koff = (lane < 16) ? 0 : 2;      // K half select (A and B symmetric)
  const int c0a  = wave * 32;
  const int c0b  = c0a + 16;
  const float* xb = x + (size_t)b * T_LEN * C_LEN;

  v8f acc00 = {};   // (t-tile 0, c-tile a)
  v8f acc01 = {};   // (t-tile 0, c-tile b)
  v8f acc10 = {};   // (t-tile 1, c-tile a)
  v8f acc11 = {};   // (t-tile 1, c-tile b)

  #pragma unroll 2
  for (int k = 0; k < T_LEN; k += 4) {
    const int ka = k + koff;
    v2f a0, a1;
    a0.x = A0[ka * 16 + lo];
    a0.y = A0[(ka + 1) * 16 + lo];
    a1.x = A1[ka * 16 + lo];
    a1.y = A1[(ka + 1) * 16 + lo];
    v2f b0, b1;
    b0.x = xb[(size_t)ka * C_LEN + c0a + lo];
    b0.y = xb[(size_t)(ka + 1) * C_LEN + c0a + lo];
    b1.x = xb[(size_t)ka * C_LEN + c0b + lo];
    b1.y = xb[(size_t)(ka + 1) * C_LEN + c0b + lo];
    acc00 = __builtin_amdgcn_wmma_f32_16x16x4_f32(false, a0, false, b0, (short)0, acc00,
                                                  false, false);
    acc01 = __builtin_amdgcn_wmma_f32_16x16x4_f32(false, a0, false, b1, (short)0, acc01,
                                                  false, false);
    acc10 = __builtin_amdgcn_wmma_f32_16x16x4_f32(false, a1, false, b0, (short)0, acc10,
                                                  false, false);
    acc11 = __builtin_amdgcn_wmma_f32_16x16x4_f32(false, a1, false, b1, (short)0, acc11,
                                                  false, false);
  }

  const int tb0 = t0 + ((lane < 16) ? 0 : 8);        // t-tile 0 base row
  const int tb1 = tb0 + 16;                          // t-tile 1 base row
  #pragma unroll
  for (int v = 0; v < 8; ++v) {
    const int ta = tb0 + v;
    const int tc = tb1 + v;
    const size_t oa0 = ((size_t)b * T_LEN + ta) * C_LEN + c0a + lo;
    const size_t oa1 = ((size_t)b * T_LEN + ta) * C_LEN + c0b + lo;
    const size_t oc0 = ((size_t)b * T_LEN + tc) * C_LEN + c0a + lo;
    const size_t oc1 = ((size_t)b * T_LEN + tc) * C_LEN + c0b + lo;
    out_tr[oa0] = acc00[v];
    out_tr[oa1] = acc01[v];
    out_tr[oc0] = acc10[v];
    out_tr[oc1] = acc11[v];
    out_sr[oa0] = xb[(size_t)ta * C_LEN + c0a + lo] - acc00[v];
    out_sr[oa1] = xb[(size_t)ta * C_LEN + c0b + lo] - acc01[v];
    out_sr[oc0] = xb[(size_t)tc * C_LEN + c0a + lo] - acc10[v];
    out_sr[oc1] = xb[(size_t)tc * C_LEN + c0b + lo] - acc11[v];
  }
}

// ---------------------------------------------------------------------------
extern "C" void kernel_launch(void* const* d_in, const int* in_sizes, int n_in,
                              void* d_out, int out_size, void* d_ws, size_t ws_size,
                              hipStream_t stream) {
  (void)in_sizes; (void)n_in; (void)out_size; (void)ws_size;
  const float* x = (const float*)d_in[0];
  float* out_sr = (float*)d_out;                                  // [B,T,C]
  float* out_tr = out_sr + (size_t)B_LEN * T_LEN * C_LEN;         // [B,T,C]
  float* W = (float*)d_ws;

  float* MLP   = W + OFF_MLP;
  float* MTR   = W + OFF_MTR;
  float* MSEAS = W + OFF_MSEAS;
  float* WCONV = W + OFF_WCONV;
  float* E     = W + OFF_E;
  float* F     = W + OFF_F;
  float* LPF   = W + OFF_LPF;
  float* S1    = W + OFF_S1;
  float* G     = W + OFF_G;
  float* H     = W + OFF_H;
  float* K     = W + OFF_K;
  float* R     = W + OFF_R;

  // --- build constant operators (deterministic, rebuilt every call) ---
  stl_build_loess<<<2, 256, 0, stream>>>(MLP,   T_LEN, 17,  0, T_LEN);
  stl_build_loess<<<2, 256, 0, stream>>>(MTR,   T_LEN, 29,  0, T_LEN);
  stl_build_loess<<<1,  64, 0, stream>>>(MSEAS, NSUB,   7, -1, EXTROWS);
  stl_build_convw<<<1, 32, 0, stream>>>(WCONV);
  stl_build_E<<<(EXTLEN * T_LEN + 255) / 256, 256, 0, stream>>>(MSEAS, E);
  stl_conv_F<<<(T_LEN * T_LEN + 255) / 256, 256, 0, stream>>>(E, WCONV, F);

  dim3 mmGrid(T_LEN / 16, T_LEN / 16), mmBlk(16, 16);
  stl_mm480<<<mmGrid, mmBlk, 0, stream>>>(MLP, F, nullptr, LPF);      // LpF = M_lp * (Λ E)
  stl_S1G<<<(T_LEN * T_LEN + 255) / 256, 256, 0, stream>>>(E, LPF, S1, G);
  stl_mm480<<<mmGrid, mmBlk, 0, stream>>>(S1, MTR, nullptr, H);       // H  = S1 * M_trend
  stl_mm480<<<mmGrid, mmBlk, 0, stream>>>(H, G, G, K);                // K  = G + H*G
  stl_mm480<<<mmGrid, mmBlk, 0, stream>>>(MTR, K, nullptr, R);        // R  = M_trend * K

  // --- main pass: trend = R x_b per batch, both outputs fused ---
  dim3 grid(T_LEN / 32, B_LEN);
  stl_trend_wmma<<<grid, 256, 0, stream>>>(x, R, out_sr, out_tr);
}